// SkipGramModel_11390253269163
// MI455X (gfx1250) — compile-verified
//
#include <hip/hip_runtime.h>
#include <math.h>

// SkipGram negative-sampling loss on MI455X (gfx1250, wave32).
//
// Roofline: ~225 MiB of random embedding-row gathers vs 0.1 GFLOP of dots:
// pure HBM-bandwidth problem (~10us floor at 23.3 TB/s). Strategy:
//  * one wave32 per 16-batch tile; the 6 dot-product groups become 6 GEMM
//    accumulations D = A(16x4 f32) x B(4x16 f32) + C via V_WMMA_F32_16X16X4_F32,
//    K=128 consumed as 32 chunks of 4.
//  * per-lane addressing for A and B is identical (lane l streams 8B chunks of
//    row (l&15) at float offset 4c + 2*(l>>4)), so every 512B row is read exactly
//    once, fully consumed, with b64 loads.
//  * the 16 wanted dots are the C-matrix diagonal (lanes 0-7 & 24-31, vgpr lane&7).
//  * deterministic reduction: shfl-xor wave reduce -> LDS -> per-block partial in
//    d_ws -> single-block ordered tree reduce writes mean(loss) to d_out.

#define EMB_DIM 128
#define NEG     5
#define NBLOCKS 256
#define BLOCK   256   // 8 waves of 32

typedef __attribute__((ext_vector_type(2))) float v2f;
typedef __attribute__((ext_vector_type(8))) float v8f;

#if __has_builtin(__builtin_amdgcn_wmma_f32_16x16x4_f32)
#define HAVE_WMMA_F32_4 1
#else
#define HAVE_WMMA_F32_4 0
#endif

__device__ __forceinline__ float clip10(float x) {
    return fminf(fmaxf(x, -10.0f), 10.0f);
}

// softplus(x) = log(1+exp(x)); inputs already clipped to [-10,10] so exp is safe.
__device__ __forceinline__ float softplus(float x) {
    return log1pf(expf(x));
}

#if HAVE_WMMA_F32_4
// Pick C[r] with r = lane&7 (diagonal element lives at VGPR index lane&7).
__device__ __forceinline__ float diag_pick(v8f c, int r) {
    float d = c[0];
    d = (r == 1) ? c[1] : d;
    d = (r == 2) ? c[2] : d;
    d = (r == 3) ? c[3] : d;
    d = (r == 4) ? c[4] : d;
    d = (r == 5) ? c[5] : d;
    d = (r == 6) ? c[6] : d;
    d = (r == 7) ? c[7] : d;
    return d;
}
#endif

__global__ __launch_bounds__(BLOCK)
void skipgram_loss_kernel(const float* __restrict__ U, const float* __restrict__ V,
                          const int* __restrict__ pos_u, const int* __restrict__ pos_v,
                          const int* __restrict__ neg_v,
                          float* __restrict__ partials, int ntiles) {
    const int lane      = threadIdx.x & 31;
    const int waveInBlk = threadIdx.x >> 5;
    const int gwave     = blockIdx.x * (BLOCK / 32) + waveInBlk;
    const int nwaves    = gridDim.x * (BLOCK / 32);

    const int m     = lane & 15;          // row of the 16-batch tile this lane serves
    const int khalf = (lane >> 4) << 1;   // 0 or 2: K offset inside each 4-wide chunk

    float lsum = 0.0f;

    // tile index is wave-uniform -> EXEC stays all-ones inside the body (WMMA req.)
    for (int tile = gwave; tile < ntiles; tile += nwaves) {
        const int b = tile * 16 + m;
        const size_t ru  = (size_t)pos_u[b];
        const size_t rv  = (size_t)pos_v[b];
        const size_t rn0 = (size_t)neg_v[b * NEG + 0];
        const size_t rn1 = (size_t)neg_v[b * NEG + 1];
        const size_t rn2 = (size_t)neg_v[b * NEG + 2];
        const size_t rn3 = (size_t)neg_v[b * NEG + 3];
        const size_t rn4 = (size_t)neg_v[b * NEG + 4];

        const float* up  = U + ru  * EMB_DIM + khalf;
        const float* vp  = V + rv  * EMB_DIM + khalf;
        const float* n0p = V + rn0 * EMB_DIM + khalf;
        const float* n1p = V + rn1 * EMB_DIM + khalf;
        const float* n2p = V + rn2 * EMB_DIM + khalf;
        const float* n3p = V + rn3 * EMB_DIM + khalf;
        const float* n4p = V + rn4 * EMB_DIM + khalf;

#if HAVE_WMMA_F32_4
        v8f cP = {}, c0 = {}, c1 = {}, c2 = {}, c3 = {}, c4 = {};
        #pragma unroll 4
        for (int c = 0; c < EMB_DIM / 4; ++c) {
            const v2f a  = *(const v2f*)(up  + 4 * c);
            const v2f bP = *(const v2f*)(vp  + 4 * c);
            const v2f b0 = *(const v2f*)(n0p + 4 * c);
            const v2f b1 = *(const v2f*)(n1p + 4 * c);
            const v2f b2 = *(const v2f*)(n2p + 4 * c);
            const v2f b3 = *(const v2f*)(n3p + 4 * c);
            const v2f b4 = *(const v2f*)(n4p + 4 * c);
            cP = __builtin_amdgcn_wmma_f32_16x16x4_f32(false, a, false, bP, (short)0, cP, false, false);
            c0 = __builtin_amdgcn_wmma_f32_16x16x4_f32(false, a, false, b0, (short)0, c0, false, false);
            c1 = __builtin_amdgcn_wmma_f32_16x16x4_f32(false, a, false, b1, (short)0, c1, false, false);
            c2 = __builtin_amdgcn_wmma_f32_16x16x4_f32(false, a, false, b2, (short)0, c2, false, false);
            c3 = __builtin_amdgcn_wmma_f32_16x16x4_f32(false, a, false, b3, (short)0, c3, false, false);
            c4 = __builtin_amdgcn_wmma_f32_16x16x4_f32(false, a, false, b4, (short)0, c4, false, false);
        }
        // Diagonal (M==N) lives in lanes 0-7 (m=lane) and 24-31 (m=lane-16), vgpr lane&7.
        const int  r     = lane & 7;
        const bool valid = (lane < 8) || (lane >= 24);
        const float sP = diag_pick(cP, r);
        const float s0 = diag_pick(c0, r);
        const float s1 = diag_pick(c1, r);
        const float s2 = diag_pick(c2, r);
        const float s3 = diag_pick(c3, r);
        const float s4 = diag_pick(c4, r);
#else
        // Fallback: paired-lane FMA dots; lanes l and l^16 each hold half of row l&15.
        float pP = 0.f, p0 = 0.f, p1 = 0.f, p2 = 0.f, p3 = 0.f, p4 = 0.f;
        #pragma unroll 4
        for (int c = 0; c < EMB_DIM / 4; ++c) {
            const v2f a  = *(const v2f*)(up  + 4 * c);
            const v2f bP = *(const v2f*)(vp  + 4 * c);
            const v2f b0 = *(const v2f*)(n0p + 4 * c);
            const v2f b1 = *(const v2f*)(n1p + 4 * c);
            const v2f b2 = *(const v2f*)(n2p + 4 * c);
            const v2f b3 = *(const v2f*)(n3p + 4 * c);
            const v2f b4 = *(const v2f*)(n4p + 4 * c);
            pP = fmaf(a[0], bP[0], fmaf(a[1], bP[1], pP));
            p0 = fmaf(a[0], b0[0], fmaf(a[1], b0[1], p0));
            p1 = fmaf(a[0], b1[0], fmaf(a[1], b1[1], p1));
            p2 = fmaf(a[0], b2[0], fmaf(a[1], b2[1], p2));
            p3 = fmaf(a[0], b3[0], fmaf(a[1], b3[1], p3));
            p4 = fmaf(a[0], b4[0], fmaf(a[1], b4[1], p4));
        }
        const float sP = pP + __shfl_xor(pP, 16, 32);
        const float s0 = p0 + __shfl_xor(p0, 16, 32);
        const float s1 = p1 + __shfl_xor(p1, 16, 32);
        const float s2 = p2 + __shfl_xor(p2, 16, 32);
        const float s3 = p3 + __shfl_xor(p3, 16, 32);
        const float s4 = p4 + __shfl_xor(p4, 16, 32);
        const bool valid = (lane < 16);
#endif
        // pos: -log_sigmoid(clip(s)) = softplus(-clip(s)); neg: softplus(clip(s))
        float contrib = softplus(-clip10(sP))
                      + softplus(clip10(s0)) + softplus(clip10(s1))
                      + softplus(clip10(s2)) + softplus(clip10(s3))
                      + softplus(clip10(s4));
        lsum += valid ? contrib : 0.0f;
    }

    // Wave32 reduction (deterministic), then ordered per-block sum.
    #pragma unroll
    for (int off = 16; off > 0; off >>= 1)
        lsum += __shfl_xor(lsum, off, 32);

    __shared__ float ssum[BLOCK / 32];
    if (lane == 0) ssum[waveInBlk] = lsum;
    __syncthreads();
    if (threadIdx.x == 0) {
        float t = 0.0f;
        #pragma unroll
        for (int i = 0; i < BLOCK / 32; ++i) t += ssum[i];
        partials[blockIdx.x] = t;   // fully overwritten every call -> no pre-zero needed
    }
}

__global__ __launch_bounds__(NBLOCKS)
void skipgram_reduce_kernel(const float* __restrict__ partials, float* __restrict__ out,
                            int n, float invB) {
    __shared__ float s[NBLOCKS];
    const int t = threadIdx.x;
    s[t] = (t < n) ? partials[t] : 0.0f;
    __syncthreads();
    #pragma unroll
    for (int off = NBLOCKS / 2; off > 0; off >>= 1) {
        if (t < off) s[t] += s[t + off];
        __syncthreads();
    }
    if (t == 0) out[0] = s[0] * invB;   // mean over batch
}

extern "C" void kernel_launch(void* const* d_in, const int* in_sizes, int n_in,
                              void* d_out, int out_size, void* d_ws, size_t ws_size,
                              hipStream_t stream) {
    const float* U     = (const float*)d_in[0];   // [1e6, 128] f32
    const float* V     = (const float*)d_in[1];   // [1e6, 128] f32
    const int*   pos_u = (const int*)d_in[2];     // [B]
    const int*   pos_v = (const int*)d_in[3];     // [B]
    const int*   neg_v = (const int*)d_in[4];     // [B, 5]

    const int B      = in_sizes[2];               // 65536
    const int ntiles = B / 16;                    // 4096 (B is a multiple of 16)

    float* partials = (float*)d_ws;               // NBLOCKS floats of scratch

    skipgram_loss_kernel<<<NBLOCKS, BLOCK, 0, stream>>>(
        U, V, pos_u, pos_v, neg_v, partials, ntiles);
    skipgram_reduce_kernel<<<1, NBLOCKS, 0, stream>>>(
        partials, (float*)d_out, NBLOCKS, 1.0f / (float)B);
}